// FCOS_6717328851129
// MI455X (gfx1250) — compile-verified
//
#include <hip/hip_runtime.h>
#include <hip/hip_bf16.h>
#include <stdint.h>

// FCOS target assignment for MI455X (gfx1250, wave32).
// cls_t: (B,P) written as float (exact for 0..80); reg_t: (B,P,4) float.
// d_out layout: [B*P] cls followed by [B*P*4] reg (flat concat in return order).

#define NUM_CLASSES 80
#define G_BOXES     64
#define INF_VAL     100000000.0f

typedef __attribute__((ext_vector_type(4))) unsigned int u32x4;
typedef __attribute__((ext_vector_type(8))) int          i32x8;
typedef __attribute__((ext_vector_type(4))) int          i32x4;

__global__ __launch_bounds__(256) void fcos_assign_kernel(
    const float* __restrict__ points,     // (P,2)
    const float* __restrict__ rranges,    // (P,2)
    const float* __restrict__ gt_bboxes,  // (B,G,4)
    const int*   __restrict__ gt_labels,  // (B,G)
    float*       __restrict__ out,        // [B*P] cls then [B*P*4] reg
    int P, int B)
{
    __shared__ float s_box[G_BOXES * 4];  // 1 KB: this batch's boxes (x,y,w,h)

    const int b = blockIdx.y;
    const int p = blockIdx.x * blockDim.x + threadIdx.x;

#if __has_builtin(__builtin_amdgcn_tensor_load_to_lds)
    // ---- TDM: DMA the 1 KB box table for batch b into LDS (one issue per block).
    if (threadIdx.x < 32) {   // whole wave 0 active -> EXEC fully set
        const uint64_t ga = (uint64_t)(uintptr_t)(gt_bboxes + (size_t)b * (G_BOXES * 4));
        const uint32_t lds_addr = (uint32_t)(uintptr_t)(void*)&s_box[0]; // flat low32 == LDS offset

        u32x4 g0;
        g0[0] = 1u;                                   // count=1 (valid), user mode
        g0[1] = lds_addr;                             // lds_addr (bytes)
        g0[2] = (uint32_t)(ga & 0xFFFFFFFFu);         // global_addr[31:0]
        g0[3] = (uint32_t)((ga >> 32) & 0x01FFFFFFu)  // global_addr[56:32]
              | (2u << 30);                           // type = 2 ("image")

        i32x8 g1;
        g1[0] = (int)(2u << 16);     // wg_mask=0 (not in cluster); data_size=2 -> 4 bytes
        g1[1] = (int)(256u << 16);   // tensor_dim0 = 256 elements (low 16 bits)
        g1[2] = (int)(1u << 16);     // tensor_dim0 hi=0 ; tensor_dim1 = 1
        g1[3] = (int)(256u << 16);   // tensor_dim1 hi=0 ; tile_dim0 = 256
        g1[4] = 1;                   // tile_dim1 = 1 ; tile_dim2 = 0 (unused)
        g1[5] = 256;                 // tensor_dim0_stride = 256 (low 32)
        g1[6] = 0;                   // stride hi / dim1_stride lo
        g1[7] = 0;                   // dim1_stride hi

        i32x4 g2 = {0, 0, 0, 0};     // no dims 2..4
        i32x4 g3 = {0, 0, 0, 0};

#if defined(__clang_major__) && (__clang_major__ >= 23)
        i32x8 g4 = {0, 0, 0, 0, 0, 0, 0, 0};
        __builtin_amdgcn_tensor_load_to_lds(g0, g1, g2, g3, g4, 0);
#else
        __builtin_amdgcn_tensor_load_to_lds(g0, g1, g2, g3, 0);
#endif
        __builtin_amdgcn_s_wait_tensorcnt(0);         // data landed in LDS
        asm volatile("" ::: "memory");
    }
#else
    // Fallback: cooperative fill (256 threads x 1 dword)
    s_box[threadIdx.x] = gt_bboxes[(size_t)b * (G_BOXES * 4) + threadIdx.x];
#endif
    __syncthreads();

    if (p >= P) return;

    const float2 pt = ((const float2*)points)[p];
    const float2 rr = ((const float2*)rranges)[p];
    const float px = pt.x, py = pt.y;
    const float lo = rr.x, hi = rr.y;

    float best  = INF_VAL;
    int   besti = 0;

    const float4* sb = (const float4*)s_box;
#pragma unroll 4
    for (int g = 0; g < G_BOXES; ++g) {
        const float4 bx = sb[g];                 // broadcast ds_load_b128
        const float dl = px - bx.x;
        const float dt = py - bx.y;
        const float dr = bx.x + bx.z - 1.0f - px;
        const float db = bx.y + bx.w - 1.0f - py;
        const float mn = fminf(fminf(dl, dt), fminf(dr, db));
        const float mx = fmaxf(fmaxf(dl, dt), fmaxf(dr, db));
        const bool valid = (mn > 0.0f) & (mx >= lo) & (mx <= hi);
        const float area = valid ? (bx.z * bx.w) : INF_VAL;
        if (area < best) { best = area; besti = g; }   // strict < == argmin first-occurrence
    }

    // Recompute the winner's distances (matches take_along_axis on reg).
    const float4 bb = sb[besti];
    float4 reg;
    reg.x = px - bb.x;                 // d_left
    reg.y = py - bb.y;                 // d_top
    reg.z = bb.x + bb.z - 1.0f - px;   // d_right
    reg.w = bb.y + bb.w - 1.0f - py;   // d_bottom

    const int cls = (best == INF_VAL) ? NUM_CLASSES
                                      : gt_labels[b * G_BOXES + besti];

    const size_t bp = (size_t)b * (size_t)P + (size_t)p;
    out[bp] = (float)cls;                                  // cls block
    float4* out_reg = (float4*)(out + (size_t)B * (size_t)P);
    out_reg[bp] = reg;                                     // 128-bit store
}

extern "C" void kernel_launch(void* const* d_in, const int* in_sizes, int n_in,
                              void* d_out, int out_size, void* d_ws, size_t ws_size,
                              hipStream_t stream) {
    const float* points  = (const float*)d_in[0];   // (P,2)
    const float* rranges = (const float*)d_in[1];   // (P,2)
    const float* boxes   = (const float*)d_in[2];   // (B,G,4)
    const int*   labels  = (const int*)d_in[3];     // (B,G)
    float* out = (float*)d_out;

    const int P = in_sizes[0] / 2;
    const int B = in_sizes[3] / G_BOXES;

    dim3 grid((P + 255) / 256, B, 1);
    fcos_assign_kernel<<<grid, dim3(256, 1, 1), 0, stream>>>(
        points, rranges, boxes, labels, out, P, B);
}